// arw_TRSNet_71201967833883
// MI455X (gfx1250) — compile-verified
//
#include <hip/hip_runtime.h>
#include <hip/hip_bf16.h>
#include <math.h>

// ---------------------------------------------------------------------------
// FoldingNet-like autoencoder on gfx1250 (MI455X), fp32 WMMA (16x16x4) path.
// ---------------------------------------------------------------------------

typedef __attribute__((ext_vector_type(2))) float v2f;
typedef __attribute__((ext_vector_type(8))) float v8f;

__device__ __forceinline__ v8f wmma4(v2f a, v2f b, v8f c) {
  // D = A(16x4,f32) * B(4x16,f32) + C(16x16,f32)
  return __builtin_amdgcn_wmma_f32_16x16x4_f32(false, a, false, b, (short)0, c,
                                               false, false);
}

// ---------------------------------------------------------------------------
// Generic GEMM: Y[M,N] = A[M,K] * W[N,K]^T (+bias[N]) (+res[M,N]) (+relu)
// Each wave computes a 16(M) x 64(N) tile; block = 4 waves = 64x64 tile.
// Hot loop is guard-free: out-of-range rows/cols are CLAMPED (they only
// pollute outputs that are never stored); only the K-tail (K%4!=0) uses
// guarded loads, executed once after the main loop.
// ---------------------------------------------------------------------------
__global__ __launch_bounds__(128) void gemm_f32_wmma(
    const float* __restrict__ A, const float* __restrict__ W,
    const float* __restrict__ bias, const float* __restrict__ res,
    float* __restrict__ Y, int M, int N, int K, int relu) {
  const int lane = threadIdx.x & 31;
  const int wave = threadIdx.x >> 5;
  const int ml = lane & 15;
  const int half = lane >> 4;
  const int tilesM = (M + 15) >> 4;
  const int tileM = blockIdx.y * 4 + wave;
  if (tileM >= tilesM) return;
  const int row = tileM * 16 + ml;
  const int n0 = blockIdx.x * 64;
  // Clamped (safe) row/col indices: garbage accumulates only into lanes whose
  // outputs are masked off at store time.
  const int rc = (row < M) ? row : 0;
  const int c0 = n0 + ml;
  const int cc0 = (c0 < N) ? c0 : 0;
  const int cc1 = (c0 + 16 < N) ? c0 + 16 : 0;
  const int cc2 = (c0 + 32 < N) ? c0 + 32 : 0;
  const int cc3 = (c0 + 48 < N) ? c0 + 48 : 0;
  const float* Ar = A + (size_t)rc * K + 2 * half;
  const float* W0 = W + (size_t)cc0 * K + 2 * half;
  const float* W1 = W + (size_t)cc1 * K + 2 * half;
  const float* W2 = W + (size_t)cc2 * K + 2 * half;
  const float* W3 = W + (size_t)cc3 * K + 2 * half;
  // Warm the streams (global_prefetch_b8).
  __builtin_prefetch(Ar, 0, 3);
  __builtin_prefetch(W0, 0, 3);
  __builtin_prefetch(W1, 0, 3);
  __builtin_prefetch(W2, 0, 3);
  __builtin_prefetch(W3, 0, 3);
  v8f acc[4];
  acc[0] = v8f{}; acc[1] = v8f{}; acc[2] = v8f{}; acc[3] = v8f{};
  const int kmain = K & ~3;
  int k0 = 0;
#pragma unroll 4
  for (; k0 + 4 <= K; k0 += 4) {
    v2f a, b0, b1, b2, b3;
    a.x = Ar[k0];  a.y = Ar[k0 + 1];
    b0.x = W0[k0]; b0.y = W0[k0 + 1];
    b1.x = W1[k0]; b1.y = W1[k0 + 1];
    b2.x = W2[k0]; b2.y = W2[k0 + 1];
    b3.x = W3[k0]; b3.y = W3[k0 + 1];
    acc[0] = wmma4(a, b0, acc[0]);
    acc[1] = wmma4(a, b1, acc[1]);
    acc[2] = wmma4(a, b2, acc[2]);
    acc[3] = wmma4(a, b3, acc[3]);
  }
  if (K & 3) {  // single guarded tail chunk (K = 258 / 259 only)
    const int ka = kmain + 2 * half;
    const bool kv0 = ka < K, kv1 = (ka + 1) < K;
    v2f a, b0, b1, b2, b3;
    a.x = kv0 ? A[(size_t)rc * K + ka] : 0.f;
    a.y = kv1 ? A[(size_t)rc * K + ka + 1] : 0.f;
    b0.x = kv0 ? W[(size_t)cc0 * K + ka] : 0.f;
    b0.y = kv1 ? W[(size_t)cc0 * K + ka + 1] : 0.f;
    b1.x = kv0 ? W[(size_t)cc1 * K + ka] : 0.f;
    b1.y = kv1 ? W[(size_t)cc1 * K + ka + 1] : 0.f;
    b2.x = kv0 ? W[(size_t)cc2 * K + ka] : 0.f;
    b2.y = kv1 ? W[(size_t)cc2 * K + ka + 1] : 0.f;
    b3.x = kv0 ? W[(size_t)cc3 * K + ka] : 0.f;
    b3.y = kv1 ? W[(size_t)cc3 * K + ka + 1] : 0.f;
    acc[0] = wmma4(a, b0, acc[0]);
    acc[1] = wmma4(a, b1, acc[1]);
    acc[2] = wmma4(a, b2, acc[2]);
    acc[3] = wmma4(a, b3, acc[3]);
  }
  // Epilogue. C layout: vgpr j, lanes 0-15 -> row j, lanes 16-31 -> row j+8.
  for (int s = 0; s < 4; ++s) {
    const int col = n0 + s * 16 + ml;
    if (col >= N) continue;
    const float bv = bias ? bias[col] : 0.f;
    for (int j = 0; j < 8; ++j) {
      const int r = tileM * 16 + j + 8 * half;
      if (r < M) {
        float v = acc[s][j] + bv;
        if (res) v += res[(size_t)r * N + col];
        if (relu) v = fmaxf(v, 0.f);
        Y[(size_t)r * N + col] = v;
      }
    }
  }
}

// ---------------------------------------------------------------------------
// Flash attention (fp32 WMMA). h=16, d=16, N=2048, B=4.
// One wave per (b, h, 16-query tile). Computes S^T = K*Q^T so that the
// softmax (over keys) reduces over the 8 C-layout VGPRs + one shfl_xor(16),
// and the online-softmax rescale is a per-lane scalar.  O^T += V^T * P^T.
// ---------------------------------------------------------------------------
__global__ __launch_bounds__(128) void flash_attn_f32(
    const float* __restrict__ Q, const float* __restrict__ Km,
    const float* __restrict__ V, float* __restrict__ O) {
  const int wid = blockIdx.x * 4 + (threadIdx.x >> 5);  // 0..8191
  const int lane = threadIdx.x & 31;
  const int ml = lane & 15;
  const int half = lane >> 4;
  const int b = wid >> 11;
  const int rem = wid & 2047;
  const int h = rem >> 7;
  const int qt = rem & 127;
  const int qrow = b * 2048 + qt * 16 + ml;
  const int hoff = h * 16;
  __builtin_prefetch(Q + (size_t)qrow * 256 + hoff, 0, 3);
  __builtin_prefetch(Km + (size_t)(b * 2048 + ml) * 256 + hoff, 0, 3);
  __builtin_prefetch(V + (size_t)(b * 2048 + ml) * 256 + hoff, 0, 3);
  float qr[8];
  for (int c = 0; c < 4; ++c)
    for (int r = 0; r < 2; ++r)
      qr[2 * c + r] =
          Q[(size_t)qrow * 256 + hoff + 4 * c + 2 * half + r] * 0.25f;
  float m_run = -1e30f, l_run = 0.f;
  v8f o = v8f{};
  for (int kt = 0; kt < 128; ++kt) {
    const int kbase = b * 2048 + kt * 16;
    v8f s = v8f{};
    for (int c = 0; c < 4; ++c) {
      v2f a, bb;
      const int d = 4 * c + 2 * half;
      a.x = Km[(size_t)(kbase + ml) * 256 + hoff + d];
      a.y = Km[(size_t)(kbase + ml) * 256 + hoff + d + 1];
      bb.x = qr[2 * c];
      bb.y = qr[2 * c + 1];
      s = wmma4(a, bb, s);  // S^T tile: lane holds keys {j+8*half}, query=ml
    }
    float mt = s[0];
    for (int j = 1; j < 8; ++j) mt = fmaxf(mt, s[j]);
    mt = fmaxf(mt, __shfl_xor(mt, 16));
    const float m_new = fmaxf(m_run, mt);
    const float fac = __expf(m_run - m_new);
    float p[8];
    float ls = 0.f;
    for (int j = 0; j < 8; ++j) {
      p[j] = __expf(s[j] - m_new);
      ls += p[j];
    }
    ls += __shfl_xor(ls, 16);
    l_run = l_run * fac + ls;
    m_run = m_new;
    for (int j = 0; j < 8; ++j) o[j] *= fac;
    // O^T += V^T * P^T, 4 K-chunks of 4 keys; P^T comes from s-registers.
    {
      const float u0 = __shfl_xor(p[2], 16), u1 = __shfl_xor(p[3], 16);
      v2f a, bb;
      a.x = V[(size_t)(kbase + 2 * half + 0) * 256 + hoff + ml];
      a.y = V[(size_t)(kbase + 2 * half + 1) * 256 + hoff + ml];
      bb.x = half ? u0 : p[0];
      bb.y = half ? u1 : p[1];
      o = wmma4(a, bb, o);
    }
    {
      const float u0 = __shfl_xor(p[6], 16), u1 = __shfl_xor(p[7], 16);
      v2f a, bb;
      a.x = V[(size_t)(kbase + 4 + 2 * half) * 256 + hoff + ml];
      a.y = V[(size_t)(kbase + 5 + 2 * half) * 256 + hoff + ml];
      bb.x = half ? u0 : p[4];
      bb.y = half ? u1 : p[5];
      o = wmma4(a, bb, o);
    }
    {
      const float u0 = __shfl_xor(p[0], 16), u1 = __shfl_xor(p[1], 16);
      v2f a, bb;
      a.x = V[(size_t)(kbase + 8 + 2 * half) * 256 + hoff + ml];
      a.y = V[(size_t)(kbase + 9 + 2 * half) * 256 + hoff + ml];
      bb.x = half ? p[2] : u0;
      bb.y = half ? p[3] : u1;
      o = wmma4(a, bb, o);
    }
    {
      const float u0 = __shfl_xor(p[4], 16), u1 = __shfl_xor(p[5], 16);
      v2f a, bb;
      a.x = V[(size_t)(kbase + 12 + 2 * half) * 256 + hoff + ml];
      a.y = V[(size_t)(kbase + 13 + 2 * half) * 256 + hoff + ml];
      bb.x = half ? p[6] : u0;
      bb.y = half ? p[7] : u1;
      o = wmma4(a, bb, o);
    }
  }
  const float inv = 1.f / l_run;
  for (int j = 0; j < 8; ++j)
    O[(size_t)qrow * 256 + hoff + j + 8 * half] = o[j] * inv;
}

// ---------------------------------------------------------------------------
// kNN (K=16) + per-point covariance -> feat12 [B*N, 12]
// ---------------------------------------------------------------------------
__global__ __launch_bounds__(256) void knn_cov(const float* __restrict__ data,
                                               float* __restrict__ feat) {
  __shared__ float pts[2048 * 3];
  const int b = blockIdx.x >> 3;
  const int chunk = blockIdx.x & 7;
  for (int i = threadIdx.x; i < 2048 * 3; i += 256)
    pts[i] = data[(size_t)b * 2048 * 3 + i];
  __syncthreads();
  const int n = chunk * 256 + threadIdx.x;
  const float px = pts[n * 3], py = pts[n * 3 + 1], pz = pts[n * 3 + 2];
  float bd[16];
  int bi[16];
  for (int i = 0; i < 16; ++i) { bd[i] = 1e30f; bi[i] = 0; }
  for (int m = 0; m < 2048; ++m) {
    const float dx = pts[m * 3] - px, dy = pts[m * 3 + 1] - py,
                dz = pts[m * 3 + 2] - pz;
    const float d2 = dx * dx + dy * dy + dz * dz;
    if (d2 < bd[15]) {
      int j = 15;
      while (j > 0 && bd[j - 1] > d2) {
        bd[j] = bd[j - 1]; bi[j] = bi[j - 1]; --j;
      }
      bd[j] = d2; bi[j] = m;
    }
  }
  float mx = 0, my = 0, mz = 0;
  for (int k = 0; k < 16; ++k) {
    mx += pts[bi[k] * 3]; my += pts[bi[k] * 3 + 1]; mz += pts[bi[k] * 3 + 2];
  }
  mx *= (1.f / 16.f); my *= (1.f / 16.f); mz *= (1.f / 16.f);
  float c[9];
  for (int i = 0; i < 9; ++i) c[i] = 0.f;
  for (int k = 0; k < 16; ++k) {
    const float fx = pts[bi[k] * 3] - mx, fy = pts[bi[k] * 3 + 1] - my,
                fz = pts[bi[k] * 3 + 2] - mz;
    c[0] += fx * fx; c[1] += fx * fy; c[2] += fx * fz;
    c[3] += fy * fx; c[4] += fy * fy; c[5] += fy * fz;
    c[6] += fz * fx; c[7] += fz * fy; c[8] += fz * fz;
  }
  float* fr = feat + (size_t)(b * 2048 + n) * 12;
  fr[0] = px; fr[1] = py; fr[2] = pz;
  for (int i = 0; i < 9; ++i) fr[3 + i] = c[i];
}

// ---------------------------------------------------------------------------
// BatchNorm stats: one block per 32-channel group; lane = adjacent channels so
// every 32-lane read is one contiguous 128B segment. 8 row-groups per block
// reduce through LDS with lane-preserving strides.
// ---------------------------------------------------------------------------
__global__ __launch_bounds__(256) void bn_stats(const float* __restrict__ X,
                                                int R, int C,
                                                float* __restrict__ mean,
                                                float* __restrict__ istd) {
  __shared__ float s1[256], s2[256];
  const int t = threadIdx.x;
  const int cl = t & 31;              // channel within group
  const int rg = t >> 5;              // row group 0..7
  const int c = blockIdx.x * 32 + cl;
  float a = 0.f, b2 = 0.f;
  for (int r = rg; r < R; r += 8) {
    const float v = X[(size_t)r * C + c];
    a += v; b2 += v * v;
  }
  s1[t] = a; s2[t] = b2;
  __syncthreads();
  for (int off = 128; off >= 32; off >>= 1) {
    if (t < off) { s1[t] += s1[t + off]; s2[t] += s2[t + off]; }
    __syncthreads();
  }
  if (t < 32) {
    const float m = s1[t] / (float)R;
    const float var = s2[t] / (float)R - m * m;
    mean[c] = m;
    istd[c] = rsqrtf(var + 1e-5f);
  }
}

__global__ void bn_apply(const float* __restrict__ X,
                         const float* __restrict__ mean,
                         const float* __restrict__ istd,
                         const float* __restrict__ g,
                         const float* __restrict__ bb, float* __restrict__ Y,
                         int R, int C, int relu) {
  const size_t i = (size_t)blockIdx.x * 256 + threadIdx.x;
  if (i < (size_t)R * C) {
    const int c = (int)(i % C);
    float v = (X[i] - mean[c]) * istd[c] * g[c] + bb[c];
    if (relu) v = fmaxf(v, 0.f);
    Y[i] = v;
  }
}

__global__ __launch_bounds__(256) void ln256(const float* __restrict__ X,
                                             const float* __restrict__ g,
                                             const float* __restrict__ bb,
                                             float* __restrict__ Y) {
  __shared__ float s1[256], s2[256];
  const int r = blockIdx.x;
  const int t = threadIdx.x;
  const float v = X[(size_t)r * 256 + t];
  s1[t] = v; s2[t] = v * v;
  __syncthreads();
  for (int off = 128; off > 0; off >>= 1) {
    if (t < off) { s1[t] += s1[t + off]; s2[t] += s2[t + off]; }
    __syncthreads();
  }
  const float m = s1[0] * (1.f / 256.f);
  const float var = s2[0] * (1.f / 256.f) - m * m;
  Y[(size_t)r * 256 + t] = (v - m) * rsqrtf(var + 1e-5f) * g[t] + bb[t];
}

__global__ __launch_bounds__(256) void maxpool_n(const float* __restrict__ X,
                                                 float* __restrict__ code) {
  const int b = blockIdx.x;
  const int c = threadIdx.x;
  float v = -1e30f;
  for (int l = 0; l < 2048; ++l)
    v = fmaxf(v, X[(size_t)(b * 2048 + l) * 256 + c]);
  code[b * 256 + c] = v;
}

// ---------------------------------------------------------------------------
// Decoder tiny MHA (seq len 4) fused with residual + LayerNorm.
// ---------------------------------------------------------------------------
__global__ __launch_bounds__(256) void dec_mha(
    const float* __restrict__ q_in, const float* __restrict__ kv_in,
    const float* __restrict__ in_w, const float* __restrict__ in_b,
    const float* __restrict__ out_w, const float* __restrict__ out_b,
    const float* __restrict__ ln_g, const float* __restrict__ ln_b,
    float* __restrict__ out) {
  __shared__ float Tq[1024], Tk[1024], Qb[1024], Kb[1024], Vb[1024], Ob[1024];
  __shared__ float Sc[256];
  __shared__ float mu[4], isd[4];
  const int t = threadIdx.x;
  for (int i = t; i < 1024; i += 256) { Tq[i] = q_in[i]; Tk[i] = kv_in[i]; }
  __syncthreads();
  for (int o = t; o < 3072; o += 256) {
    const int which = o >> 10;
    const int s = (o >> 8) & 3;
    const int e = o & 255;
    const float* src = (which == 0) ? Tq : Tk;
    const float* wr = in_w + (size_t)(which * 256 + e) * 256;
    float acc = in_b[which * 256 + e];
    for (int i = 0; i < 256; ++i) acc += src[s * 256 + i] * wr[i];
    float* dst = (which == 0) ? Qb : (which == 1) ? Kb : Vb;
    dst[s * 256 + e] = acc;
  }
  __syncthreads();
  {  // scores: t = h*16 + qi*4 + kj
    const int hh = t >> 4, qi = (t >> 2) & 3, kj = t & 3;
    float acc = 0.f;
    for (int d = 0; d < 16; ++d)
      acc += Qb[qi * 256 + hh * 16 + d] * Kb[kj * 256 + hh * 16 + d];
    Sc[t] = acc * 0.25f;
  }
  __syncthreads();
  if (t < 64) {
    const int base = t * 4;
    float m = Sc[base];
    for (int j = 1; j < 4; ++j) m = fmaxf(m, Sc[base + j]);
    float ssum = 0.f;
    for (int j = 0; j < 4; ++j) {
      const float e = __expf(Sc[base + j] - m);
      Sc[base + j] = e; ssum += e;
    }
    for (int j = 0; j < 4; ++j) Sc[base + j] /= ssum;
  }
  __syncthreads();
  for (int o = t; o < 1024; o += 256) {
    const int qi = o >> 8, f = o & 255, hh = f >> 4;
    float acc = 0.f;
    for (int j = 0; j < 4; ++j) acc += Sc[hh * 16 + qi * 4 + j] * Vb[j * 256 + f];
    Ob[o] = acc;
  }
  __syncthreads();
  for (int o = t; o < 1024; o += 256) {
    const int s = o >> 8, e = o & 255;
    const float* wr = out_w + (size_t)e * 256;
    float acc = out_b[e];
    for (int i = 0; i < 256; ++i) acc += Ob[s * 256 + i] * wr[i];
    Qb[o] = Tq[o] + acc;  // residual, reuse Qb as pre-LN buffer
  }
  __syncthreads();
  if (t < 4) {
    float a = 0.f, b2 = 0.f;
    for (int i = 0; i < 256; ++i) {
      const float v = Qb[t * 256 + i];
      a += v; b2 += v * v;
    }
    const float m = a * (1.f / 256.f);
    mu[t] = m;
    isd[t] = rsqrtf(b2 * (1.f / 256.f) - m * m + 1e-5f);
  }
  __syncthreads();
  for (int o = t; o < 1024; o += 256) {
    const int s = o >> 8, e = o & 255;
    out[o] = (Qb[o] - mu[s]) * isd[s] * ln_g[e] + ln_b[e];
  }
}

// ---------------------------------------------------------------------------
// Folding-grid generation, concat, output gather
// ---------------------------------------------------------------------------
__global__ void grid_gen(float* __restrict__ g) {
  const int i = blockIdx.x * 256 + threadIdx.x;
  if (i < 2500) g[i] = -40.f + (float)(i % 50) * (80.f / 49.f);
  else if (i < 5000) g[i] = -60.f + (float)((i - 2500) / 50) * (120.f / 49.f);
}

// mode 0: grids = grid2 [2,2500] broadcast over b (2 channels)
// mode 1: grids = recon1 [B*M, 3] (3 channels)
__global__ void cat_rows(const float* __restrict__ grids,
                         const float* __restrict__ cw, float* __restrict__ Y,
                         int gch, int mode) {
  const int C = gch + 256;
  const size_t i = (size_t)blockIdx.x * 256 + threadIdx.x;
  if (i >= (size_t)10000 * C) return;
  const int r = (int)(i / C);
  const int c = (int)(i % C);
  const int b = r / 2500;
  const int m = r % 2500;
  float v;
  if (c < gch)
    v = (mode == 0) ? grids[c * 2500 + m] : grids[(size_t)r * 3 + c];
  else
    v = cw[b * 256 + (c - gch)];
  Y[i] = v;
}

// Deterministic stand-in for jax.random.permutation(key(1), 2500)[:2048]
// (a true permutation of Z_2500 via a coprime stride; same gather structure).
__global__ void gather_out(const float* __restrict__ R2,
                           float* __restrict__ out) {
  const int i = blockIdx.x * 256 + threadIdx.x;
  if (i >= 4 * 2048 * 3) return;
  const int c = i % 3;
  const int n = (i / 3) % 2048;
  const int b = i / (3 * 2048);
  const int perm = (int)(((long long)n * 1117LL + 613LL) % 2500LL);
  out[i] = R2[(size_t)(b * 2500 + perm) * 3 + c];
}

// ---------------------------------------------------------------------------
// Parameter index table (alphabetical dict flattening: data, num_nodes, params)
// ---------------------------------------------------------------------------
enum {
  IN_DATA = 0, IN_NUMNODES,
  P_BN1_B, P_BN1_G, P_BN2_B, P_BN2_G, P_BN3_B, P_BN3_G, P_BN4_B, P_BN4_G,
  P_CONV1_B, P_CONV1_W, P_CONV2_B, P_CONV2_W, P_CONV3_B, P_CONV3_W,
  P_CONV4_B, P_CONV4_W,
  P_DEC_CATTN_INB, P_DEC_CATTN_INW, P_DEC_CATTN_OUTB, P_DEC_CATTN_OUTW,
  P_DEC_LIN1_B, P_DEC_LIN1_W, P_DEC_LIN2_B, P_DEC_LIN2_W,
  P_DEC_LN1_B, P_DEC_LN1_G, P_DEC_LN2_B, P_DEC_LN2_G, P_DEC_LN3_B, P_DEC_LN3_G,
  P_DEC_SATTN_INB, P_DEC_SATTN_INW, P_DEC_SATTN_OUTB, P_DEC_SATTN_OUTW,
  P_ENC_ATTN_INB, P_ENC_ATTN_INW, P_ENC_ATTN_OUTB, P_ENC_ATTN_OUTW,
  P_ENC_LIN1_B, P_ENC_LIN1_W, P_ENC_LIN2_B, P_ENC_LIN2_W,
  P_ENC_LN1_B, P_ENC_LN1_G, P_ENC_LN2_B, P_ENC_LN2_G,
  P_F1_BN1_B, P_F1_BN1_G, P_F1_BN2_B, P_F1_BN2_G,
  P_F1_C1_B, P_F1_C1_W, P_F1_C2_B, P_F1_C2_W, P_F1_C3_B, P_F1_C3_W,
  P_F2_BN1_B, P_F2_BN1_G, P_F2_BN2_B, P_F2_BN2_G,
  P_F2_C1_B, P_F2_C1_W, P_F2_C2_B, P_F2_C2_W, P_F2_C3_B, P_F2_C3_W
};

// Workspace layout (float offsets). Requires ws_size >= ~119 MB.
static constexpr size_t OFF_FEAT = 0;          // 98304
static constexpr size_t OFF_B1 = 100000;       // conv1-out / attn-out
static constexpr size_t OFF_B2 = 2200000;      // conv2-out / proj+res
static constexpr size_t OFF_B3 = 4300000;      // conv3-out x0 / conv4-out
static constexpr size_t OFF_B4 = 6400000;      // Q / x1 (post-ln1)
static constexpr size_t OFF_B5 = 8500000;      // K / ffn-out
static constexpr size_t OFF_B6 = 10600000;     // V / x2 (post-ln2)
static constexpr size_t OFF_B7 = 12700000;     // FFN mid (16.8M) / fold region
static constexpr size_t OFF_CAT1 = OFF_B7;
static constexpr size_t OFF_F1A = OFF_B7 + 2600000;
static constexpr size_t OFF_F1B = OFF_B7 + 5200000;
static constexpr size_t OFF_CAT2 = OFF_B7 + 7800000;
static constexpr size_t OFF_F2A = OFF_B7 + 10400000;
static constexpr size_t OFF_F2B = OFF_B7 + 13000000;
static constexpr size_t OFF_R1 = OFF_B7 + 15600000;
static constexpr size_t OFF_R2 = OFF_B7 + 15700000;
static constexpr size_t OFF_MEAN = 29500000;
static constexpr size_t OFF_ISTD = 29503000;
static constexpr size_t OFF_CODE = 29506000;
static constexpr size_t OFF_GRID = 29508000;
static constexpr size_t OFF_T1 = 29514000;
static constexpr size_t OFF_T2 = 29516000;
static constexpr size_t OFF_DMID = 29518000;
static constexpr size_t OFF_DFFN = 29527000;
static constexpr size_t OFF_TF = 29529000;

static void gemm(hipStream_t st, const float* A, const float* W,
                 const float* bias, const float* res, float* Y, int M, int N,
                 int K, int relu) {
  dim3 g((N + 63) / 64, (M + 63) / 64);
  gemm_f32_wmma<<<g, 128, 0, st>>>(A, W, bias, res, Y, M, N, K, relu);
}

static void bn(hipStream_t st, const float* X, float* Y, const float* g,
               const float* b, float* mean, float* istd, int R, int C,
               int relu) {
  bn_stats<<<C / 32, 256, 0, st>>>(X, R, C, mean, istd);
  const size_t tot = (size_t)R * C;
  bn_apply<<<(unsigned)((tot + 255) / 256), 256, 0, st>>>(X, mean, istd, g, b,
                                                          Y, R, C, relu);
}

extern "C" void kernel_launch(void* const* d_in, const int* in_sizes, int n_in,
                              void* d_out, int out_size, void* d_ws,
                              size_t ws_size, hipStream_t stream) {
  (void)in_sizes; (void)n_in; (void)out_size; (void)ws_size;
  const float* data = (const float*)d_in[IN_DATA];
  auto P = [&](int i) { return (const float*)d_in[i]; };
  float* ws = (float*)d_ws;

  float* feat = ws + OFF_FEAT;
  float* b1 = ws + OFF_B1;
  float* b2 = ws + OFF_B2;
  float* x0 = ws + OFF_B3;
  float* qb = ws + OFF_B4;
  float* kb = ws + OFF_B5;
  float* vb = ws + OFF_B6;
  float* mid = ws + OFF_B7;
  float* mean = ws + OFF_MEAN;
  float* istd = ws + OFF_ISTD;
  float* code = ws + OFF_CODE;
  float* grd = ws + OFF_GRID;
  float* t1 = ws + OFF_T1;
  float* t2 = ws + OFF_T2;
  float* dmid = ws + OFF_DMID;
  float* dffn = ws + OFF_DFFN;
  float* tf = ws + OFF_TF;

  const int BN = 8192;   // B*N rows
  const int BM = 10000;  // B*M rows

  // ---- encode: knn + cov + conv stack ----
  knn_cov<<<32, 256, 0, stream>>>(data, feat);
  gemm(stream, feat, P(P_CONV1_W), P(P_CONV1_B), nullptr, b1, BN, 128, 12, 0);
  bn(stream, b1, b1, P(P_BN1_G), P(P_BN1_B), mean, istd, BN, 128, 1);
  gemm(stream, b1, P(P_CONV2_W), P(P_CONV2_B), nullptr, b2, BN, 128, 128, 0);
  bn(stream, b2, b2, P(P_BN2_G), P(P_BN2_B), mean, istd, BN, 128, 1);
  gemm(stream, b2, P(P_CONV3_W), P(P_CONV3_B), nullptr, x0, BN, 256, 128, 0);
  bn(stream, x0, x0, P(P_BN3_G), P(P_BN3_B), mean, istd, BN, 256, 1);

  // ---- encoder layer: MHA ----
  const float* inw = P(P_ENC_ATTN_INW);
  const float* inb = P(P_ENC_ATTN_INB);
  gemm(stream, x0, inw, inb, nullptr, qb, BN, 256, 256, 0);
  gemm(stream, x0, inw + 256 * 256, inb + 256, nullptr, kb, BN, 256, 256, 0);
  gemm(stream, x0, inw + 512 * 256, inb + 512, nullptr, vb, BN, 256, 256, 0);
  flash_attn_f32<<<2048, 128, 0, stream>>>(qb, kb, vb, b1);
  gemm(stream, b1, P(P_ENC_ATTN_OUTW), P(P_ENC_ATTN_OUTB), x0, b2, BN, 256,
       256, 0);
  float* x1 = qb;  // Q dead, reuse
  ln256<<<BN, 256, 0, stream>>>(b2, P(P_ENC_LN1_G), P(P_ENC_LN1_B), x1);
  // ---- encoder FFN ----
  gemm(stream, x1, P(P_ENC_LIN1_W), P(P_ENC_LIN1_B), nullptr, mid, BN, 2048,
       256, 1);
  float* fo = kb;  // K dead
  gemm(stream, mid, P(P_ENC_LIN2_W), P(P_ENC_LIN2_B), x1, fo, BN, 256, 2048, 0);
  float* x2 = vb;  // V dead
  ln256<<<BN, 256, 0, stream>>>(fo, P(P_ENC_LN2_G), P(P_ENC_LN2_B), x2);
  // ---- conv4 + bn4 + maxpool -> code ----
  gemm(stream, x2, P(P_CONV4_W), P(P_CONV4_B), nullptr, x0, BN, 256, 256, 0);
  bn(stream, x0, b1, P(P_BN4_G), P(P_BN4_B), mean, istd, BN, 256, 0);
  maxpool_n<<<4, 256, 0, stream>>>(b1, code);

  // ---- decoder (seq len 4) ----
  dec_mha<<<1, 256, 0, stream>>>(code, code, P(P_DEC_SATTN_INW),
                                 P(P_DEC_SATTN_INB), P(P_DEC_SATTN_OUTW),
                                 P(P_DEC_SATTN_OUTB), P(P_DEC_LN1_G),
                                 P(P_DEC_LN1_B), t1);
  dec_mha<<<1, 256, 0, stream>>>(t1, code, P(P_DEC_CATTN_INW),
                                 P(P_DEC_CATTN_INB), P(P_DEC_CATTN_OUTW),
                                 P(P_DEC_CATTN_OUTB), P(P_DEC_LN2_G),
                                 P(P_DEC_LN2_B), t2);
  gemm(stream, t2, P(P_DEC_LIN1_W), P(P_DEC_LIN1_B), nullptr, dmid, 4, 2048,
       256, 1);
  gemm(stream, dmid, P(P_DEC_LIN2_W), P(P_DEC_LIN2_B), t2, dffn, 4, 256, 2048,
       0);
  ln256<<<4, 256, 0, stream>>>(dffn, P(P_DEC_LN3_G), P(P_DEC_LN3_B), tf);

  // ---- folding ----
  float* cat1 = ws + OFF_CAT1;
  float* f1a = ws + OFF_F1A;
  float* f1b = ws + OFF_F1B;
  float* cat2 = ws + OFF_CAT2;
  float* f2a = ws + OFF_F2A;
  float* f2b = ws + OFF_F2B;
  float* r1 = ws + OFF_R1;
  float* r2 = ws + OFF_R2;
  grid_gen<<<(5000 + 255) / 256, 256, 0, stream>>>(grd);
  cat_rows<<<(unsigned)(((size_t)BM * 258 + 255) / 256), 256, 0, stream>>>(
      grd, tf, cat1, 2, 0);
  gemm(stream, cat1, P(P_F1_C1_W), P(P_F1_C1_B), nullptr, f1a, BM, 256, 258, 0);
  bn(stream, f1a, f1a, P(P_F1_BN1_G), P(P_F1_BN1_B), mean, istd, BM, 256, 1);
  gemm(stream, f1a, P(P_F1_C2_W), P(P_F1_C2_B), nullptr, f1b, BM, 256, 256, 0);
  bn(stream, f1b, f1b, P(P_F1_BN2_G), P(P_F1_BN2_B), mean, istd, BM, 256, 1);
  gemm(stream, f1b, P(P_F1_C3_W), P(P_F1_C3_B), nullptr, r1, BM, 3, 256, 0);
  cat_rows<<<(unsigned)(((size_t)BM * 259 + 255) / 256), 256, 0, stream>>>(
      r1, tf, cat2, 3, 1);
  gemm(stream, cat2, P(P_F2_C1_W), P(P_F2_C1_B), nullptr, f2a, BM, 256, 259, 0);
  bn(stream, f2a, f2a, P(P_F2_BN1_G), P(P_F2_BN1_B), mean, istd, BM, 256, 1);
  gemm(stream, f2a, P(P_F2_C2_W), P(P_F2_C2_B), nullptr, f2b, BM, 256, 256, 0);
  bn(stream, f2b, f2b, P(P_F2_BN2_G), P(P_F2_BN2_B), mean, istd, BM, 256, 1);
  gemm(stream, f2b, P(P_F2_C3_W), P(P_F2_C3_B), nullptr, r2, BM, 3, 256, 0);

  gather_out<<<(4 * 2048 * 3 + 255) / 256, 256, 0, stream>>>(r2, (float*)d_out);
}